// Lloss_84172769067233
// MI455X (gfx1250) — compile-verified
//
#include <hip/hip_runtime.h>
#include <stdint.h>

// ---------------- compile-time geometry ----------------
#define NSTOCKS   4000
#define ROWBYTES  (NSTOCKS * 4u)     // 16000 B per row
#define ROWS      128                // rows per block == threads per block
#define CCOLS     128                // columns per full chunk (512 B / row = 1 wave of b128)
#define NFULL     31                 // 31 full chunks cover cols 0..3967
#define REMC      32                 // remainder chunk: cols 3968..3999 (128 B / row = 1 wave of b32)
#define PITCH     132                // floats per LDS row (+4 pad -> minimal 2-phase b128 bank access)
#define TILEF     (ROWS * PITCH)     // floats per LDS buffer
#define SHBYTES   (2 * TILEF * 4 + ROWS * 8)   // double-buffered tiles + double reduction array

#define LOG2E_F   1.44269504088896340736f

// ---------------- CDNA5 async global->LDS helpers ----------------
__device__ __forceinline__ void async_ld_b128(unsigned ldsoff, unsigned goff,
                                              unsigned long long base) {
    asm volatile("global_load_async_to_lds_b128 %0, %1, %2"
                 :: "v"(ldsoff), "v"(goff), "s"(base) : "memory");
}
__device__ __forceinline__ void async_ld_b32(unsigned ldsoff, unsigned goff,
                                             unsigned long long base) {
    asm volatile("global_load_async_to_lds_b32 %0, %1, %2"
                 :: "v"(ldsoff), "v"(goff), "s"(base) : "memory");
}
template <int N> __device__ __forceinline__ void wait_asynccnt() {
    asm volatile("s_wait_asynccnt %0" :: "n"(N) : "memory");
}

// ---------------- streaming logsumexp (base-2 domain) ----------------
// Per-row state: M = running max of y = f*log2e, S = sum 2^(y-M) in [1, N].
// accA accumulates M per element; log2 terms batched as log2(prod S) per quad.
__device__ __forceinline__ void lse_step(float y, float& M, float& S, float& P,
                                         float& accA, float& accY) {
    float d = y - M;
    if (d > 0.0f) {                                   // rare: new running max
        S = S * __builtin_amdgcn_exp2f(-d) + 1.0f;
        M = y;
    } else {
        S = S + __builtin_amdgcn_exp2f(d);
    }
    P *= S;
    accA += M;
    accY += y;
}

// Process one float4 (scan order: w, z, y, x). Branch is hoisted to quad level:
// the no-new-max fast path is fully branchless with 4 independent v_exp_f32.
__device__ __forceinline__ void lse_quad(float4 v, float& M, float& S,
                                         float& accA, float& accY) {
    float y3 = v.w * LOG2E_F;
    float y2 = v.z * LOG2E_F;
    float y1 = v.y * LOG2E_F;
    float y0 = v.x * LOG2E_F;
    float P = 1.0f;
    float mq = fmaxf(fmaxf(y3, y2), fmaxf(y1, y0));
    if (mq <= M) {                                    // common path: M unchanged
        float e3 = __builtin_amdgcn_exp2f(y3 - M);
        float e2 = __builtin_amdgcn_exp2f(y2 - M);
        float e1 = __builtin_amdgcn_exp2f(y1 - M);
        float e0 = __builtin_amdgcn_exp2f(y0 - M);
        S += e3; P *= S;
        S += e2; P *= S;
        S += e1; P *= S;
        S += e0; P *= S;
        accA += 4.0f * M;                             // exact in fp32
        accY += (y3 + y2) + (y1 + y0);
    } else {                                          // rare (~ln N per row): careful path
        lse_step(y3, M, S, P, accA, accY);
        lse_step(y2, M, S, P, accA, accY);
        lse_step(y1, M, S, P, accA, accY);
        lse_step(y0, M, S, P, accA, accY);
    }
    accA += __builtin_amdgcn_logf(P);                 // v_log_f32 (log2): 1 per 4 elements
}

// ---------------- main kernel: one thread per row ----------------
__global__ __launch_bounds__(ROWS) void listmle_rows(const float* __restrict__ f,
                                                     double* __restrict__ bsum) {
    extern __shared__ float lds[];                    // [2*TILEF floats][ROWS doubles]
    const int t     = threadIdx.x;
    const int lane  = t & 31;
    const int wrow0 = t & ~31;                        // first row (in block) owned by this wave
    const unsigned long long gbase = (unsigned long long)(uintptr_t)f;
    const unsigned blockBase = (unsigned)blockIdx.x * (unsigned)ROWS * ROWBYTES;
    const unsigned ldsBase   = (unsigned)(uintptr_t)lds;   // low 32 bits = LDS offset

    const unsigned gWave = blockBase + (unsigned)wrow0 * ROWBYTES;       // global base, wave rows
    const unsigned lWave = ldsBase + (unsigned)wrow0 * (PITCH * 4u);     // LDS base, wave rows

    // ---- issue remainder chunk (cols 3968..3999) into buffer 0 (coalesced b32, in-bounds)
    {
        unsigned g = gWave + (unsigned)(NFULL * CCOLS) * 4u + (unsigned)lane * 4u;
        unsigned l = lWave + (unsigned)lane * 4u;
        #pragma unroll 4
        for (int j = 0; j < 32; ++j) {
            async_ld_b32(l, g, gbase);
            g += ROWBYTES; l += PITCH * 4u;
        }
    }
    // ---- issue full chunk 30 into buffer 1 (one wave-wide b128 per row = 512 B coalesced)
    {
        unsigned g = gWave + 30u * (CCOLS * 4u) + (unsigned)lane * 16u;
        unsigned l = lWave + (unsigned)(TILEF * 4) + (unsigned)lane * 16u;
        #pragma unroll 4
        for (int j = 0; j < 32; ++j) {
            async_ld_b128(l, g, gbase);
            g += ROWBYTES; l += PITCH * 4u;
        }
    }

    float M = -__builtin_huge_valf();
    float S = 0.0f;
    double lseSum = 0.0, ySum = 0.0;

    // ---- process remainder chunk first (scan runs from column N-1 downward)
    wait_asynccnt<32>();                              // in-order completion: rem chunk landed
    {
        float accA = 0.0f, accY = 0.0f;
        const float* rowp = lds + t * PITCH;
        #pragma unroll
        for (int q = REMC / 4 - 1; q >= 0; --q) {
            float4 v = *(const float4*)(rowp + 4 * q);          // ds_load_b128
            lse_quad(v, M, S, accA, accY);
        }
        lseSum += (double)accA; ySum += (double)accY;
    }

    // ---- double-buffered pipeline over full chunks k = 30 .. 0
    int bsel = 1;
    for (int k = 30; k >= 0; --k) {
        if (k > 0) {
            unsigned g = gWave + (unsigned)(k - 1) * (CCOLS * 4u) + (unsigned)lane * 16u;
            unsigned l = lWave + (unsigned)(bsel ^ 1) * (unsigned)(TILEF * 4)
                       + (unsigned)lane * 16u;
            #pragma unroll 4
            for (int j = 0; j < 32; ++j) {
                async_ld_b128(l, g, gbase);
                g += ROWBYTES; l += PITCH * 4u;
            }
            wait_asynccnt<32>();     // <=32 outstanding => chunk k fully landed (in-order)
        } else {
            wait_asynccnt<0>();
        }
        float accA = 0.0f, accY = 0.0f;
        const float* rowp = lds + bsel * TILEF + t * PITCH;
        #pragma unroll 4
        for (int q = CCOLS / 4 - 1; q >= 0; --q) {
            float4 v = *(const float4*)(rowp + 4 * q);
            lse_quad(v, M, S, accA, accY);
        }
        lseSum += (double)accA; ySum += (double)accY;
        bsel ^= 1;
    }

    // ---- deterministic block reduction of per-row (base-2) totals
    double* red = (double*)(lds + 2 * TILEF);
    red[t] = lseSum - ySum;
    __syncthreads();
    for (int s2 = ROWS / 2; s2 > 0; s2 >>= 1) {
        if (t < s2) red[t] += red[t + s2];
        __syncthreads();
    }
    if (t == 0) bsum[blockIdx.x] = red[0];
}

// ---------------- deterministic final reduce: 128 block sums -> scalar ----------------
__global__ __launch_bounds__(256) void listmle_finalize(const double* __restrict__ bsum,
                                                        int nb, float* __restrict__ out,
                                                        double scale) {
    __shared__ double red[256];
    int t = threadIdx.x;
    red[t] = (t < nb) ? bsum[t] : 0.0;
    __syncthreads();
    for (int s = 128; s > 0; s >>= 1) {
        if (t < s) red[t] += red[t + s];
        __syncthreads();
    }
    if (t == 0) out[0] = (float)(red[0] * scale);   // scale = ln2 / B (base-2 -> natural log, mean)
}

extern "C" void kernel_launch(void* const* d_in, const int* in_sizes, int n_in,
                              void* d_out, int out_size, void* d_ws, size_t ws_size,
                              hipStream_t stream) {
    const float* f = (const float*)d_in[0];
    const int B = in_sizes[0] / NSTOCKS;          // 16384
    const int nblocks = B / ROWS;                 // 128
    double* bsum = (double*)d_ws;                 // 128 doubles of scratch

    hipFuncSetAttribute(reinterpret_cast<const void*>(listmle_rows),
                        hipFuncAttributeMaxDynamicSharedMemorySize, SHBYTES);

    listmle_rows<<<nblocks, ROWS, SHBYTES, stream>>>(f, bsum);

    const double scale = 0.69314718055994530942 / (double)B;   // ln2 / B
    listmle_finalize<<<1, 256, 0, stream>>>(bsum, nblocks, (float*)d_out, scale);
}